// DenseGNOBlock_73830487818864
// MI455X (gfx1250) — compile-verified
//
#include <hip/hip_runtime.h>
#include <math.h>

typedef __attribute__((ext_vector_type(2))) float v2f;
typedef __attribute__((ext_vector_type(8))) float v8f;

#define BATCH 4
#define SEQ   8192
#define CH    64
#define LSTR  68          // padded LDS row stride (floats) -> conflict-free frag reads
#define SCALE (1.0f/65536.0f)   // 1/(sqrt(64)*8192)

__device__ __forceinline__ v8f wmma_f32(v2f a, v2f b, v8f c) {
  // V_WMMA_F32_16X16X4_F32: D = A(16x4) * B(4x16) + C(16x16), f32 throughout
  return __builtin_amdgcn_wmma_f32_16x16x4_f32(false, a, false, b, (short)0, c, false, false);
}

__device__ __forceinline__ v8f v8_splat(float s) {
  v8f r;
#pragma unroll
  for (int i = 0; i < 8; ++i) r[i] = s;
  return r;
}

// ---------------- pass 0: zero the KtV workspace ----------------
__global__ void zero_ws(float* __restrict__ p, int n) {
  int i = blockIdx.x * blockDim.x + threadIdx.x;
  if (i < n) p[i] = 0.0f;
}

// ---------------- pass 1: KtV[b] = sum_n k[b,n,:]^T v[b,n,:] ----------------
// grid: 64 blocks (16 per batch) x 128 threads (4 waves); 8 tiles of 16 rows per wave
__global__ void __launch_bounds__(128) pass1_ktv(
    const float* __restrict__ x,
    const float* __restrict__ Wk, const float* __restrict__ bk,
    const float* __restrict__ Wv, const float* __restrict__ bv,
    float* __restrict__ ktv) {
  __shared__ float sAcc[CH * CH];        // 16 KB block-level KtV accumulator
  __shared__ float sK[4][16 * LSTR];     // per-wave k tile
  __shared__ float sV[4][16 * LSTR];     // per-wave v tile

  const int tid  = threadIdx.x;
  const int wave = tid >> 5;
  const int lane = tid & 31;
  const int l16  = lane & 15;
  const int sel  = (lane >> 4) << 1;     // 0 for lanes 0-15, 2 for lanes 16-31
  const int roff = sel << 2;             // 0 or 8 (D-layout row offset)

  for (int i = tid; i < CH * CH; i += 128) sAcc[i] = 0.0f;
  __syncthreads();

  const int batch = blockIdx.x >> 4;
  const int sub   = blockIdx.x & 15;

  float bkv[4], bvv[4];
#pragma unroll
  for (int j = 0; j < 4; ++j) { bkv[j] = bk[16 * j + l16]; bvv[j] = bv[16 * j + l16]; }

  float* myK = sK[wave];
  float* myV = sV[wave];

  for (int it = 0; it < 8; ++it) {
    const int  tile = sub * 32 + wave * 8 + it;             // 0..511 within batch
    const long row0 = (long)batch * SEQ + (long)tile * 16;
    const float* xrow = x + (row0 + l16) * CH;

    // x A-fragments (16x4 tiles along K)
    v2f xa[16];
#pragma unroll
    for (int c = 0; c < 16; ++c) xa[c] = *(const v2f*)(xrow + 4 * c + sel);

    // k = x*Wk^T + bk ; v = x*Wv^T + bv ; stage D-layout -> LDS row-major
#pragma unroll
    for (int j = 0; j < 4; ++j) {
      v8f ak = v8_splat(bkv[j]);
      v8f av = v8_splat(bvv[j]);
#pragma unroll
      for (int c = 0; c < 16; ++c) {
        v2f bf = *(const v2f*)(Wk + (16 * j + l16) * CH + 4 * c + sel);
        ak = wmma_f32(xa[c], bf, ak);
      }
#pragma unroll
      for (int c = 0; c < 16; ++c) {
        v2f bf = *(const v2f*)(Wv + (16 * j + l16) * CH + 4 * c + sel);
        av = wmma_f32(xa[c], bf, av);
      }
#pragma unroll
      for (int r = 0; r < 8; ++r) {
        myK[(r + roff) * LSTR + 16 * j + l16] = ak[r];
        myV[(r + roff) * LSTR + 16 * j + l16] = av[r];
      }
    }
    asm volatile("s_wait_dscnt 0x0" ::: "memory");   // LDS in-order per wave; fence vs compiler too

    // KtV(64x64) += k^T(64x16) * v(16x64)
#pragma unroll
    for (int i = 0; i < 4; ++i) {
#pragma unroll
      for (int j = 0; j < 4; ++j) {
        v8f acc = v8_splat(0.0f);
#pragma unroll
        for (int c = 0; c < 4; ++c) {
          v2f af, bf;
          af.x = myK[(4 * c + sel)     * LSTR + 16 * i + l16];
          af.y = myK[(4 * c + sel + 1) * LSTR + 16 * i + l16];
          bf.x = myV[(4 * c + sel)     * LSTR + 16 * j + l16];
          bf.y = myV[(4 * c + sel + 1) * LSTR + 16 * j + l16];
          acc = wmma_f32(af, bf, acc);
        }
#pragma unroll
        for (int r = 0; r < 8; ++r)
          atomicAdd(&sAcc[(16 * i + r + roff) * CH + 16 * j + l16], acc[r]);
      }
    }
    asm volatile("s_wait_dscnt 0x0" ::: "memory");
  }

  __syncthreads();
  float* dst = ktv + batch * CH * CH;
  for (int i = tid; i < CH * CH; i += 128) atomicAdd(&dst[i], sAcc[i]);
}

// ---------------- pass 2: out = gelu(x*Ww^T + bw + (x*Wq^T + bq) * KtV * scale) ----------------
// grid: 256 blocks x 256 threads (8 waves); one 16-row tile per wave (2048 tiles total)
__global__ void __launch_bounds__(256) pass2_out(
    const float* __restrict__ x,
    const float* __restrict__ Wq, const float* __restrict__ bq,
    const float* __restrict__ Ww, const float* __restrict__ bw,
    const float* __restrict__ ktv,
    float* __restrict__ out) {
  __shared__ float sQ[8][16 * LSTR];     // per-wave q staging (D-layout -> A-layout)

  const int tid  = threadIdx.x;
  const int wave = tid >> 5;
  const int lane = tid & 31;
  const int l16  = lane & 15;
  const int sel  = (lane >> 4) << 1;
  const int roff = sel << 2;

  const int  tile  = blockIdx.x * 8 + wave;   // 0..2047
  const int  batch = tile >> 9;               // 512 tiles per batch
  const long row0  = (long)tile * 16;         // contiguous across batches
  const float* xrow = x + (row0 + l16) * CH;

  v2f xa[16];
#pragma unroll
  for (int c = 0; c < 16; ++c) xa[c] = *(const v2f*)(xrow + 4 * c + sel);

  float* myQ = sQ[wave];

  // q tile (scale folded in), staged to LDS
#pragma unroll
  for (int j = 0; j < 4; ++j) {
    v8f acc = v8_splat(bq[16 * j + l16]);
#pragma unroll
    for (int c = 0; c < 16; ++c) {
      v2f bf = *(const v2f*)(Wq + (16 * j + l16) * CH + 4 * c + sel);
      acc = wmma_f32(xa[c], bf, acc);
    }
#pragma unroll
    for (int r = 0; r < 8; ++r)
      myQ[(r + roff) * LSTR + 16 * j + l16] = acc[r] * SCALE;
  }
  asm volatile("s_wait_dscnt 0x0" ::: "memory");

  v2f qa[16];
#pragma unroll
  for (int c = 0; c < 16; ++c)
    qa[c] = *(const v2f*)(myQ + l16 * LSTR + 4 * c + sel);

  const float* M = ktv + batch * CH * CH;

#pragma unroll
  for (int j = 0; j < 4; ++j) {
    v8f acc = v8_splat(bw[16 * j + l16]);
#pragma unroll
    for (int c = 0; c < 16; ++c) {            // w = x*Ww^T + bw
      v2f bf = *(const v2f*)(Ww + (16 * j + l16) * CH + 4 * c + sel);
      acc = wmma_f32(xa[c], bf, acc);
    }
#pragma unroll
    for (int c = 0; c < 16; ++c) {            // += q_scaled * KtV
      v2f bf;
      bf.x = M[(4 * c + sel)     * CH + 16 * j + l16];
      bf.y = M[(4 * c + sel + 1) * CH + 16 * j + l16];
      acc = wmma_f32(qa[c], bf, acc);
    }
#pragma unroll
    for (int r = 0; r < 8; ++r) {             // exact GELU + store (D-layout)
      float xv = acc[r];
      float g  = 0.5f * xv * (1.0f + erff(xv * 0.70710678118654752f));
      out[(row0 + r + roff) * CH + 16 * j + l16] = g;
    }
  }
}

extern "C" void kernel_launch(void* const* d_in, const int* in_sizes, int n_in,
                              void* d_out, int out_size, void* d_ws, size_t ws_size,
                              hipStream_t stream) {
  const float* x  = (const float*)d_in[0];
  const float* Wq = (const float*)d_in[1];
  const float* bq = (const float*)d_in[2];
  const float* Wk = (const float*)d_in[3];
  const float* bk = (const float*)d_in[4];
  const float* Wv = (const float*)d_in[5];
  const float* bv = (const float*)d_in[6];
  const float* Ww = (const float*)d_in[7];
  const float* bw = (const float*)d_in[8];
  float* out = (float*)d_out;
  float* ktv = (float*)d_ws;                 // needs BATCH*64*64*4 = 64 KB of scratch

  zero_ws<<<(BATCH * CH * CH + 255) / 256, 256, 0, stream>>>(ktv, BATCH * CH * CH);
  pass1_ktv<<<64, 128, 0, stream>>>(x, Wk, bk, Wv, bv, ktv);
  pass2_out<<<256, 256, 0, stream>>>(x, Wq, bq, Ww, bw, ktv, out);
}